// GridSecurityGNN_87282325389840
// MI455X (gfx1250) — compile-verified
//
#include <hip/hip_runtime.h>
#include <math.h>

#define N_NODES 50000
#define N_EDGES 800000
#define IN_DIM 32
#define HID 64
#define OUT_DIM 4
#define HEADS 4
#define LAYERS 3
#define N_GRAPHS 50

typedef float v2f __attribute__((ext_vector_type(2)));
typedef float v8f __attribute__((ext_vector_type(8)));

// ---------------------------------------------------------------------------
// fp32 WMMA GEMM, compile-time K and N: C[M,N] = A[M,K] @ B[K,N] (+ bias[N]).
// One wave computes a 16x16 tile of C via V_WMMA_F32_16X16X4_F32, K stepped
// by 4 and fully unrolled (K/4 back-to-back WMMAs). Block = 128 threads =
// 4 waves; wave w handles ntile = blockIdx.y*4 + w. M multiple of 16.
//
// Fragment layouts (ISA 7.12.2):
//   A 16x4 : lane -> row m = lane&15; VGPR{0,1} = K {kh, kh+1}, kh = 2*(lane>>4)
//   B 4x16 : lane -> col n = lane&15; VGPR{0,1} = K {kh, kh+1}
//   C 16x16: VGPR v -> row m = v + 8*(lane>>4), col n = lane&15
// ---------------------------------------------------------------------------
template <int K, int N, bool HAS_BIAS>
__global__ void __launch_bounds__(128)
gemm_wmma_f32_t(const float* __restrict__ A, const float* __restrict__ B,
                const float* __restrict__ bias, float* __restrict__ C) {
  const int lane  = threadIdx.x & 31;
  const int wave  = threadIdx.x >> 5;
  const int mtile = blockIdx.x;
  const int ntile = blockIdx.y * 4 + wave;
  const int mbase = mtile * 16;
  const int nbase = ntile * 16;
  const int mr    = lane & 15;   // A row / B col / C col within tile
  const int half  = lane >> 4;   // 0 or 1
  const int kh    = half * 2;    // lanes 0-15 -> K{0,1}; lanes 16-31 -> K{2,3}

  const float* ap = A + (size_t)(mbase + mr) * K + kh;        // 8B aligned
  const float* bp = B + (size_t)kh * N + (nbase + mr);

  v8f acc = {};
#pragma unroll
  for (int k0 = 0; k0 < K; k0 += 4) {
    v2f a = *(const v2f*)ap;     // A[m, k0+kh], A[m, k0+kh+1]
    v2f b;
    b.x = bp[0];                 // B[k0+kh,   n]
    b.y = bp[N];                 // B[k0+kh+1, n]
    acc = __builtin_amdgcn_wmma_f32_16x16x4_f32(
        /*neg_a=*/false, a, /*neg_b=*/false, b,
        /*c_mod=*/(short)0, acc, /*reuse_a=*/false, /*reuse_b=*/false);
    ap += 4;
    bp += 4 * N;
  }

  const int n = nbase + mr;
  const float bv = HAS_BIAS ? bias[n] : 0.0f;
  float* cp = C + (size_t)(mbase + 8 * half) * N + n;
#pragma unroll
  for (int v = 0; v < 8; ++v) {
    cp[(size_t)v * N] = acc[v] + bv;
  }
}

// ---------------------------------------------------------------------------
// Per-(node,head) attention scores: sc_src[n,h] = <xt[n,h,:], a_src[h,:]>
// ---------------------------------------------------------------------------
__global__ void k_scores(const float* __restrict__ xt,
                         const float* __restrict__ a_src,
                         const float* __restrict__ a_dst,
                         float* __restrict__ sc_src,
                         float* __restrict__ sc_dst) {
  int i = blockIdx.x * blockDim.x + threadIdx.x;  // over N*HEADS
  if (i >= N_NODES * HEADS) return;
  int n = i >> 2, h = i & 3;
  const float* row = xt + (size_t)n * (HEADS * HID) + h * HID;
  const float* as = a_src + h * HID;
  const float* ad = a_dst + h * HID;
  float ss = 0.f, sd = 0.f;
#pragma unroll 8
  for (int c = 0; c < HID; ++c) {
    float v = row[c];
    ss += v * as[c];
    sd += v * ad[c];
  }
  sc_src[i] = ss;
  sc_dst[i] = sd;
}

// ---------------------------------------------------------------------------
// Init per-layer reduction buffers.
// ---------------------------------------------------------------------------
__global__ void k_init(float* __restrict__ agg, float* __restrict__ emax,
                       float* __restrict__ denom) {
  int i = blockIdx.x * blockDim.x + threadIdx.x;
  if (i < N_NODES * HID) agg[i] = 0.0f;
  if (i < N_NODES * HEADS) {
    emax[i]  = -3.0e38f;
    denom[i] = 0.0f;
  }
}

__global__ void k_zero(float* __restrict__ p, int n) {
  int i = blockIdx.x * blockDim.x + threadIdx.x;
  if (i < n) p[i] = 0.0f;
}

// float atomic max via sign-split int/uint trick (works for mixed signs)
__device__ __forceinline__ void atomicMaxF(float* addr, float val) {
  if (val >= 0.0f)
    atomicMax((int*)addr, __float_as_int(val));
  else
    atomicMin((unsigned int*)addr, __float_as_uint(val));
}

__device__ __forceinline__ float lrelu(float x) {
  return x > 0.0f ? x : 0.2f * x;
}

// ---------------------------------------------------------------------------
// Pass 1: segment max of leaky-relu edge scores per (dst, head).
// Edge e >= N_EDGES is the self-loop for node (e - N_EDGES).
// ---------------------------------------------------------------------------
__global__ void k_edge_max(const int* __restrict__ srcIdx,
                           const int* __restrict__ dstIdx,
                           const float* __restrict__ sc_src,
                           const float* __restrict__ sc_dst,
                           float* __restrict__ emax) {
  const int ETOT = N_EDGES + N_NODES;
  int i = blockIdx.x * blockDim.x + threadIdx.x;  // over ETOT*HEADS
  if (i >= ETOT * HEADS) return;
  int e = i >> 2, h = i & 3;
  int s = (e < N_EDGES) ? srcIdx[e] : (e - N_EDGES);
  int d = (e < N_EDGES) ? dstIdx[e] : (e - N_EDGES);
  float sc = lrelu(sc_src[s * HEADS + h] + sc_dst[d * HEADS + h]);
  atomicMaxF(&emax[d * HEADS + h], sc);
}

// ---------------------------------------------------------------------------
// Pass 2: segment sum of exp(score - max).
// ---------------------------------------------------------------------------
__global__ void k_edge_denom(const int* __restrict__ srcIdx,
                             const int* __restrict__ dstIdx,
                             const float* __restrict__ sc_src,
                             const float* __restrict__ sc_dst,
                             const float* __restrict__ emax,
                             float* __restrict__ denom) {
  const int ETOT = N_EDGES + N_NODES;
  int i = blockIdx.x * blockDim.x + threadIdx.x;
  if (i >= ETOT * HEADS) return;
  int e = i >> 2, h = i & 3;
  int s = (e < N_EDGES) ? srcIdx[e] : (e - N_EDGES);
  int d = (e < N_EDGES) ? dstIdx[e] : (e - N_EDGES);
  float sc = lrelu(sc_src[s * HEADS + h] + sc_dst[d * HEADS + h]);
  float ex = __expf(sc - emax[d * HEADS + h]);
  atomicAdd(&denom[d * HEADS + h], ex);
}

// ---------------------------------------------------------------------------
// Pass 3: weighted message aggregation with the head-mean folded in:
//   agg[d, c] += 0.25 * sum_h alpha[e,h] * xt[s, h, c]
// 64 consecutive threads per edge -> coalesced gather + coalesced atomics.
// ---------------------------------------------------------------------------
__global__ void k_message(const int* __restrict__ srcIdx,
                          const int* __restrict__ dstIdx,
                          const float* __restrict__ sc_src,
                          const float* __restrict__ sc_dst,
                          const float* __restrict__ emax,
                          const float* __restrict__ denom,
                          const float* __restrict__ xt,
                          float* __restrict__ agg) {
  const int ETOT = N_EDGES + N_NODES;
  int t = blockIdx.x * blockDim.x + threadIdx.x;  // over ETOT*HID
  int e = t >> 6;
  int c = t & 63;
  if (e >= ETOT) return;
  int s = (e < N_EDGES) ? srcIdx[e] : (e - N_EDGES);
  int d = (e < N_EDGES) ? dstIdx[e] : (e - N_EDGES);
  const float* xrow = xt + (size_t)s * (HEADS * HID);
  float acc = 0.0f;
#pragma unroll
  for (int h = 0; h < HEADS; ++h) {
    float sc = lrelu(sc_src[s * HEADS + h] + sc_dst[d * HEADS + h]);
    float alpha = __expf(sc - emax[d * HEADS + h]) /
                  (denom[d * HEADS + h] + 1e-16f);
    acc += xrow[h * HID + c] * alpha;
  }
  atomicAdd(&agg[(size_t)d * HID + c], 0.25f * acc);
}

// ---------------------------------------------------------------------------
// Fused: bias + BatchNorm(eval) + ReLU + residual, in-place on h.
// ---------------------------------------------------------------------------
__global__ void k_finalize(float* __restrict__ h, const float* __restrict__ agg,
                           const float* __restrict__ gb,
                           const float* __restrict__ bn_g,
                           const float* __restrict__ bn_b,
                           const float* __restrict__ bn_m,
                           const float* __restrict__ bn_v) {
  int i = blockIdx.x * blockDim.x + threadIdx.x;
  if (i >= N_NODES * HID) return;
  int c = i & 63;
  float val = agg[i] + gb[c];
  val = (val - bn_m[c]) * rsqrtf(bn_v[c] + 1e-5f) * bn_g[c] + bn_b[c];
  val = val > 0.0f ? val : 0.0f;
  h[i] = h[i] + val;
}

// ---------------------------------------------------------------------------
// Global mean pool (accumulate).
// ---------------------------------------------------------------------------
__global__ void k_pool(const int* __restrict__ batch,
                       const float* __restrict__ h,
                       float* __restrict__ sums, float* __restrict__ cnt) {
  int i = blockIdx.x * blockDim.x + threadIdx.x;
  if (i >= N_NODES * HID) return;
  int n = i >> 6, c = i & 63;
  int b = batch[n];
  atomicAdd(&sums[b * HID + c], h[i]);
  if (c == 0) atomicAdd(&cnt[b], 1.0f);
}

// ---------------------------------------------------------------------------
// Head: g = sums/cnt; out = relu(g@fc1+b1)@fc2+b2. One block per graph.
// ---------------------------------------------------------------------------
__global__ void k_head(const float* __restrict__ sums,
                       const float* __restrict__ cnt,
                       const float* __restrict__ fc1W,
                       const float* __restrict__ fc1b,
                       const float* __restrict__ fc2W,
                       const float* __restrict__ fc2b,
                       float* __restrict__ out) {
  __shared__ float g[HID];
  __shared__ float f1[HID];
  int b = blockIdx.x;
  int t = threadIdx.x;
  float c = cnt[b];
  c = c > 1.0f ? c : 1.0f;
  g[t] = sums[b * HID + t] / c;
  __syncthreads();
  float acc = fc1b[t];
#pragma unroll 8
  for (int k = 0; k < HID; ++k) acc += g[k] * fc1W[k * HID + t];
  f1[t] = acc > 0.0f ? acc : 0.0f;
  __syncthreads();
  if (t < OUT_DIM) {
    float o = fc2b[t];
#pragma unroll 8
    for (int k = 0; k < HID; ++k) o += f1[k] * fc2W[k * OUT_DIM + t];
    out[b * OUT_DIM + t] = o;
  }
}

// ---------------------------------------------------------------------------
extern "C" void kernel_launch(void* const* d_in, const int* in_sizes, int n_in,
                              void* d_out, int out_size, void* d_ws, size_t ws_size,
                              hipStream_t stream) {
  const float* x       = (const float*)d_in[0];
  const int*   eidx    = (const int*)d_in[1];       // [2, E]
  const int*   batch   = (const int*)d_in[2];
  const float* in_W    = (const float*)d_in[3];
  const float* in_b    = (const float*)d_in[4];
  const float* gat_W   = (const float*)d_in[5];     // [L, 64, 256]
  const float* att_src = (const float*)d_in[6];     // [L, 4, 64]
  const float* att_dst = (const float*)d_in[7];
  const float* gat_b   = (const float*)d_in[8];     // [L, 64]
  const float* bn_g    = (const float*)d_in[9];
  const float* bn_b    = (const float*)d_in[10];
  const float* bn_m    = (const float*)d_in[11];
  const float* bn_v    = (const float*)d_in[12];
  const float* fc1_W   = (const float*)d_in[13];
  const float* fc1_b   = (const float*)d_in[14];
  const float* fc2_W   = (const float*)d_in[15];
  const float* fc2_b   = (const float*)d_in[16];

  const int* srcIdx = eidx;
  const int* dstIdx = eidx + N_EDGES;

  // Workspace layout (floats); total ~80 MB
  float* ws    = (float*)d_ws;
  float* h     = ws;                                  // N*64
  float* xt    = h     + (size_t)N_NODES * HID;       // N*256
  float* agg   = xt    + (size_t)N_NODES * HEADS*HID; // N*64
  float* scs   = agg   + (size_t)N_NODES * HID;       // N*4
  float* scd   = scs   + (size_t)N_NODES * HEADS;     // N*4
  float* emax  = scd   + (size_t)N_NODES * HEADS;     // N*4
  float* denom = emax  + (size_t)N_NODES * HEADS;     // N*4
  float* sums  = denom + (size_t)N_NODES * HEADS;     // 50*64
  float* cnt   = sums  + N_GRAPHS * HID;              // 50

  const int ETOT = N_EDGES + N_NODES;
  const int MT   = N_NODES / 16;  // 3125

  // 1) h = x @ in_W + in_b  (M=50000, K=32, N=64; 4 n-tiles = 4 waves)
  gemm_wmma_f32_t<IN_DIM, HID, true><<<dim3(MT, 1), 128, 0, stream>>>(
      x, in_W, in_b, h);

  for (int l = 0; l < LAYERS; ++l) {
    const float* Wl  = gat_W   + (size_t)l * HID * HEADS * HID;
    const float* asl = att_src + (size_t)l * HEADS * HID;
    const float* adl = att_dst + (size_t)l * HEADS * HID;

    // 2a) xt = h @ W  (M=50000, K=64, N=256; 16 n-tiles = 4 blocks.y x 4 waves)
    gemm_wmma_f32_t<HID, HEADS * HID, false><<<dim3(MT, 4), 128, 0, stream>>>(
        h, Wl, nullptr, xt);
    // 2b) attention score dot products
    k_scores<<<(N_NODES * HEADS + 255) / 256, 256, 0, stream>>>(xt, asl, adl,
                                                                scs, scd);
    // 2c) init reductions
    k_init<<<(N_NODES * HID + 255) / 256, 256, 0, stream>>>(agg, emax, denom);
    // 2d) segment max
    k_edge_max<<<(ETOT * HEADS + 255) / 256, 256, 0, stream>>>(srcIdx, dstIdx,
                                                               scs, scd, emax);
    // 2e) segment sum of exp
    k_edge_denom<<<(ETOT * HEADS + 255) / 256, 256, 0, stream>>>(
        srcIdx, dstIdx, scs, scd, emax, denom);
    // 2f) message aggregation (head mean folded in)
    k_message<<<(ETOT * HID + 255) / 256, 256, 0, stream>>>(
        srcIdx, dstIdx, scs, scd, emax, denom, xt, agg);
    // 2g) bias + BN + ReLU + residual
    k_finalize<<<(N_NODES * HID + 255) / 256, 256, 0, stream>>>(
        h, agg, gat_b + l * HID, bn_g + l * HID, bn_b + l * HID,
        bn_m + l * HID, bn_v + l * HID);
  }

  // 3) global mean pool + MLP head
  k_zero<<<(N_GRAPHS * HID + N_GRAPHS + 255) / 256, 256, 0, stream>>>(
      sums, N_GRAPHS * HID + N_GRAPHS);
  k_pool<<<(N_NODES * HID + 255) / 256, 256, 0, stream>>>(batch, h, sums, cnt);
  k_head<<<N_GRAPHS, HID, 0, stream>>>(sums, cnt, fc1_W, fc1_b, fc2_W, fc2_b,
                                       (float*)d_out);
}